// NeuralDictionaryV6_19430432047762
// MI455X (gfx1250) — compile-verified
//
#include <hip/hip_runtime.h>

typedef float v2f __attribute__((ext_vector_type(2)));
typedef float v4f __attribute__((ext_vector_type(4)));
typedef float v8f __attribute__((ext_vector_type(8)));
typedef int   v4i __attribute__((vector_size(16)));   // matches builtin param type

#define NROWS 262144
#define DDIM  1024
#define DVDIM 1024

// ---------------- Pass 1: logits = keys @ query, via V_WMMA_F32_16X16X4_F32 ----
// 4 waves/block, 16 rows/wave, per-wave double-buffered LDS tile, no barriers.
#define P1_WAVES          4
#define P1_ROWS_PER_WAVE  16
#define P1_ROWS_PER_BLOCK 64
#define P1_GRID           (NROWS / P1_ROWS_PER_BLOCK)   // 4096
#define NWAVE1            (P1_GRID * P1_WAVES)          // 16384
#define KCHUNK            64
#define NCHUNK            (DDIM / KCHUNK)               // 16
#define PSTR              68   // padded row stride (floats): 68 mod 64 = 4 -> conflict-free b64 A-reads

#if __has_builtin(__builtin_amdgcn_global_load_async_to_lds_b128) && \
    __has_builtin(__builtin_amdgcn_s_wait_asynccnt)
#define HAVE_ASYNC_LDS 1
#else
#define HAVE_ASYNC_LDS 0
#endif

typedef __attribute__((address_space(1))) v4i* gptr_t;  // global (AS1) int4*
typedef __attribute__((address_space(3))) v4i* lptr_t;  // LDS (AS3) int4*

// Stage 16 rows x 64 floats of keys into a per-wave LDS tile.
// 8 async copies: each covers 2 rows x 256B, fully coalesced, no VGPR
// round-trip, tracked by ASYNCcnt. Lane (hi,lm) handles row r+hi, cols lm*4..+3.
__device__ __forceinline__ void stage_chunk(const float* __restrict__ keys,
                                            int rowBase, int hi, int lm,
                                            int kc, float* buf)
{
#if HAVE_ASYNC_LDS
  #pragma unroll
  for (int r = 0; r < P1_ROWS_PER_WAVE; r += 2) {
    const float* g = keys + (size_t)(rowBase + r + hi) * DDIM + kc + lm * 4;
    float*       l = buf + (r + hi) * PSTR + lm * 4;
    // flat global == AS(1) bitwise; flat->AS(3) is low-32-bit truncation (ISA 10.2).
    __builtin_amdgcn_global_load_async_to_lds_b128(
        (gptr_t)(unsigned long long)g,
        (lptr_t)(unsigned int)(unsigned long long)l,
        0, 0);
  }
#else
  #pragma unroll
  for (int r = 0; r < P1_ROWS_PER_WAVE; r += 2) {
    const v4f* g = (const v4f*)(keys + (size_t)(rowBase + r + hi) * DDIM + kc) + lm;
    v4f v = __builtin_nontemporal_load(g);
    *(v4f*)&buf[(r + hi) * PSTR + lm * 4] = v;
  }
#endif
}

__device__ __forceinline__ void wait_async_le8(void) {
#if HAVE_ASYNC_LDS
  __builtin_amdgcn_s_wait_asynccnt(8);
#endif
}
__device__ __forceinline__ void wait_async_0(void) {
#if HAVE_ASYNC_LDS
  __builtin_amdgcn_s_wait_asynccnt(0);
#endif
}

__global__ __launch_bounds__(128) void k1_logits(
    const float* __restrict__ query, const float* __restrict__ keys,
    float* __restrict__ logits, float* __restrict__ wavemax)
{
  __shared__ float tile[P1_WAVES][2][P1_ROWS_PER_WAVE * PSTR];  // ~34.8 KB

  const int tid  = threadIdx.x;
  const int wave = tid >> 5;
  const int lane = tid & 31;
  const int hi   = lane >> 4;   // 0: lanes 0-15, 1: lanes 16-31
  const int lm   = lane & 15;
  const int rowBase = blockIdx.x * P1_ROWS_PER_BLOCK + wave * P1_ROWS_PER_WAVE;

  float* buf0 = &tile[wave][0][0];
  float* buf1 = &tile[wave][1][0];

  v8f acc = {};  // 16x16 f32 accumulator (8 VGPRs)

  stage_chunk(keys, rowBase, hi, lm, 0, buf0);  // prime buffer 0

  for (int c = 0; c < NCHUNK; ++c) {
    float* cur = (c & 1) ? buf1 : buf0;
    float* nxt = (c & 1) ? buf0 : buf1;
    const int kc = c * KCHUNK;
    if (c + 1 < NCHUNK) {
      stage_chunk(keys, rowBase, hi, lm, kc + KCHUNK, nxt);  // overlap DMA
      wait_async_le8();   // oldest 8 (chunk c) done; async completes in-order
    } else {
      wait_async_0();
    }
    // 16 WMMAs: A = keys tile (ISA 16x4 f32 A-layout from LDS),
    // B = query chunk replicated across all 16 columns.
    #pragma unroll
    for (int k0 = 0; k0 < KCHUNK; k0 += 4) {
      v2f a = *(const v2f*)&cur[lm * PSTR + k0 + 2 * hi];  // ds_load_b64
      v2f q01 = *(const v2f*)(query + kc + k0);
      v2f q23 = *(const v2f*)(query + kc + k0 + 2);
      v2f b;
      b.x = hi ? q01.y : q01.x;   // vgpr0: row k=0 (lanes 0-15) / k=1 (lanes 16-31)
      b.y = hi ? q23.y : q23.x;   // vgpr1: row k=2 / k=3
      acc = __builtin_amdgcn_wmma_f32_16x16x4_f32(
          false, a, false, b, (short)0, acc, false, false);
    }
  }

  // D columns identical: lanes 0-15 hold rows 0-7 in acc[0..7], lanes 16-31 rows 8-15.
  float lmax = acc[0];
  #pragma unroll
  for (int v = 1; v < 8; ++v) lmax = fmaxf(lmax, acc[v]);
  float wmax = fmaxf(lmax, __shfl_xor(lmax, 16, 32));

  if (lm == 0) {
    float* dst = logits + rowBase + hi * 8;
    #pragma unroll
    for (int v = 0; v < 8; ++v) dst[v] = acc[v];
  }
  if (lane == 0) wavemax[blockIdx.x * P1_WAVES + wave] = wmax;
}

// ---------------- Pass 2: global max over 16384 wave maxima ------------------
__global__ __launch_bounds__(256) void k2_max(const float* __restrict__ wm,
                                              float* __restrict__ Mout)
{
  __shared__ float s[256];
  const int tid = threadIdx.x;
  float m = -3.4e38f;
  for (int i = tid; i < NWAVE1; i += 256) m = fmaxf(m, wm[i]);
  s[tid] = m; __syncthreads();
  for (int o = 128; o > 0; o >>= 1) {
    if (tid < o) s[tid] = fmaxf(s[tid], s[tid + o]);
    __syncthreads();
  }
  if (tid == 0) Mout[0] = s[0];
}

// ---------------- Pass 3: partial out_b = sum_i exp(l_i - M) * values[i] -----
#define NB3 1024
#define R3  (NROWS / NB3)   // 256 rows per block

__global__ __launch_bounds__(256) void k3_values(
    const float* __restrict__ logits, const float* __restrict__ values,
    const float* __restrict__ Mptr,
    float* __restrict__ pout, float* __restrict__ psum)
{
  const int b = blockIdx.x, t = threadIdx.x;
  const float M = Mptr[0];
  const int row0 = b * R3;
  const float* vptr = values + (size_t)row0 * DVDIM + t * 4;  // thread t owns cols 4t..4t+3
  v4f acc = {};
  float ssum = 0.f;
  #pragma unroll 8
  for (int i = 0; i < R3; ++i) {
    float w = __expf(logits[row0 + i] - M);   // uniform load + v_exp
    v4f v = __builtin_nontemporal_load((const v4f*)vptr);
    acc += w * v;
    ssum += w;
    vptr += DVDIM;
  }
  *(v4f*)(pout + (size_t)b * DVDIM + t * 4) = acc;
  if (t == 0) psum[b] = ssum;
}

// ---------------- Pass 4: out[j] = sum_b pout[b][j] / sum_b psum[b] ----------
__global__ __launch_bounds__(256) void k4_final(
    const float* __restrict__ pout, const float* __restrict__ psum,
    float* __restrict__ out)
{
  __shared__ float s[256];
  const int tid = threadIdx.x;
  float ss = 0.f;
  for (int i = tid; i < NB3; i += 256) ss += psum[i];
  s[tid] = ss; __syncthreads();
  for (int o = 128; o > 0; o >>= 1) {
    if (tid < o) s[tid] += s[tid + o];
    __syncthreads();
  }
  const float S = s[0];
  const int j = blockIdx.x * 256 + tid;
  float a = 0.f;
  #pragma unroll 8
  for (int b = 0; b < NB3; ++b) a += pout[(size_t)b * DVDIM + j];
  out[j] = a / S;
}

// ---------------- launcher ---------------------------------------------------
extern "C" void kernel_launch(void* const* d_in, const int* in_sizes, int n_in,
                              void* d_out, int out_size, void* d_ws, size_t ws_size,
                              hipStream_t stream)
{
  const float* query  = (const float*)d_in[0];   // [1024]
  const float* keys   = (const float*)d_in[1];   // [262144, 1024]
  const float* values = (const float*)d_in[2];   // [262144, 1024]
  float* out = (float*)d_out;                    // [1024] f32

  // workspace layout (floats), all 16B aligned: total ~5.4 MB
  float* wsf      = (float*)d_ws;
  float* logits   = wsf;                          // NROWS
  float* wavemax  = logits + NROWS;               // NWAVE1 (16384)
  float* Mval     = wavemax + NWAVE1;             // 1 (+pad)
  float* psum     = Mval + 64;                    // NB3
  float* pout     = psum + NB3;                   // NB3 * DVDIM

  k1_logits<<<P1_GRID, 128, 0, stream>>>(query, keys, logits, wavemax);
  k2_max  <<<1,       256, 0, stream>>>(wavemax, Mval);
  k3_values<<<NB3,    256, 0, stream>>>(logits, values, Mval, pout, psum);
  k4_final<<<DVDIM / 256, 256, 0, stream>>>(pout, psum, out);
}